// RBFKernel_65481071405061
// MI455X (gfx1250) — compile-verified
//
#include <hip/hip_runtime.h>

typedef __attribute__((ext_vector_type(2))) float v2f;
typedef __attribute__((ext_vector_type(8))) float v8f;

#define NROWS 8192
#define DIM   64
#define TILE  128   // output tile per 256-thread block (8 waves x 16 rows)

// ptrtoint of an addrspace(3) pointer yields the wave-relative LDS byte offset,
// which is what async-to-LDS instructions take in their VDST operand.
__device__ inline unsigned lds_offset(const void* p) {
    return (unsigned)(unsigned long long)(__attribute__((address_space(3))) const void*)p;
}

__global__ __launch_bounds__(256) void rbf_wmma_f32(const float* __restrict__ x,
                                                    const float* __restrict__ y,
                                                    float* __restrict__ out) {
    __shared__ float ytile[TILE * DIM];   // 32 KB, XOR-swizzled rows
    __shared__ float ysq[TILE];           // column norms

    const int tid     = threadIdx.x;
    const int tileRow = blockIdx.y * TILE;
    const int tileCol = blockIdx.x * TILE;

    const int wave = tid >> 5;   // 0..7 : which 16-row strip of the tile
    const int lane = tid & 31;
    const int lo   = lane & 15;  // M (A) / N (B,C,D) index within fragment
    const int hi   = lane >> 4;  // selects K pair (A/B) or M+8 half (C/D)

    // --- async-stage the 128x64 y tile into LDS (2048 float4, 8 per thread).
    // XOR swizzle: element (row, d) lives at dword 64*row + (d ^ (4*(row&15)))
    // -> B-fragment ds_load_b64 reads hit all 64 banks conflict-free.
    {
        const float* ybase = y + (size_t)tileCol * DIM;
        #pragma unroll
        for (int i = 0; i < 8; ++i) {
            const int idx = tid + i * 256;      // float4 index within tile
            const int row = idx >> 4;           // 16 float4 per row
            const int c4  = (idx & 15) * 4;     // dword within row
            const unsigned goff = (unsigned)((row * DIM + c4) * 4);
            const unsigned loff = lds_offset(&ytile[row * DIM + (c4 ^ (4 * (row & 15)))]);
            asm volatile("global_load_async_to_lds_b128 %0, %1, %2"
                         :: "v"(loff), "v"(goff), "s"(ybase) : "memory");
        }
    }

    // --- A fragments straight from global (L2-hot, loaded once, reused 8x);
    // overlaps with the in-flight async staging.
    // ISA 16x4 f32 A layout: lanes 0-15 hold (M=lane, K=k..k+1),
    // lanes 16-31 hold (M=lane-16, K=k+2..k+3).
    const float* arow = x + (size_t)(tileRow + wave * 16 + lo) * DIM + hi * 2;
    v2f a[16];
    #pragma unroll
    for (int kk = 0; kk < 16; ++kk) {
        a[kk] = *reinterpret_cast<const v2f*>(arow + kk * 4);
    }

    // --- row norms from registers: this lane holds half of K for row `lo`;
    // lane^16 holds the other half. Then ds_bpermute fans out the 8 norms
    // each lane needs for its C/D rows (M = r + 8*hi). Once per wave: cheap.
    float xpart = 0.0f;
    #pragma unroll
    for (int kk = 0; kk < 16; ++kk) xpart += a[kk].x * a[kk].x + a[kk].y * a[kk].y;
    const float xsq = xpart + __shfl_xor(xpart, 16, 32);
    float xn[8];
    #pragma unroll
    for (int r = 0; r < 8; ++r) {
        xn[r] = __int_as_float(
            __builtin_amdgcn_ds_bpermute(4 * (r + 8 * hi), __float_as_int(xsq)));
    }

    asm volatile("s_wait_asynccnt 0x0" ::: "memory");
    __syncthreads();

    // --- cooperative column norms from the staged tile: one norm per thread,
    // once per block (keeps the k-loop free of VALU clutter).
    if (tid < TILE) {
        const float* row = &ytile[tid * DIM];
        const int sw = 4 * (tid & 15);
        float s = 0.0f;
        #pragma unroll
        for (int c = 0; c < DIM; c += 4) {
            const float4 v = *reinterpret_cast<const float4*>(row + (c ^ sw));
            s += v.x * v.x + v.y * v.y + v.z * v.z + v.w * v.w;
        }
        ysq[tid] = s;
    }
    __syncthreads();

    // --- sweep 8 column tiles; B fragments from LDS; 16 WMMAs (K=4) each.
    // Inner loop is pure ds_load + v_wmma.
    #pragma unroll
    for (int ct = 0; ct < 8; ++ct) {
        const int rl = ct * 16 + lo;            // y row == output column (local)
        const float* bbase = &ytile[rl * DIM];
        const int sw = 4 * lo;                  // swizzle key (rl & 15 == lo)

        v8f acc = {};
        #pragma unroll
        for (int kk = 0; kk < 16; ++kk) {
            v2f b = *reinterpret_cast<const v2f*>(bbase + ((kk * 4 + hi * 2) ^ sw));
            acc = __builtin_amdgcn_wmma_f32_16x16x4_f32(
                /*neg_a=*/false, a[kk], /*neg_b=*/false, b,
                /*c_mod=*/(short)0, acc, /*reuse_a=*/false, /*reuse_b=*/false);
        }

        // --- fused epilogue: dist = xn + yn - 2*cross; clamp; exp(-dist) ---
        const float yn  = ysq[rl];
        const int   col = tileCol + rl;
        #pragma unroll
        for (int r = 0; r < 8; ++r) {
            const int row = tileRow + wave * 16 + r + hi * 8;
            float dist = fmaxf(xn[r] + yn - 2.0f * acc[r], 0.0f);
            out[(size_t)row * NROWS + col] = __expf(-dist);   // GAMMA = 1.0
        }
    }
}

extern "C" void kernel_launch(void* const* d_in, const int* in_sizes, int n_in,
                              void* d_out, int out_size, void* d_ws, size_t ws_size,
                              hipStream_t stream) {
    (void)in_sizes; (void)n_in; (void)d_ws; (void)ws_size; (void)out_size;
    const float* x = (const float*)d_in[0];
    const float* y = (const float*)d_in[1];
    float* out = (float*)d_out;

    dim3 grid(NROWS / TILE, NROWS / TILE);  // 64 x 64 blocks
    dim3 block(256);                        // 8 waves (wave32)
    rbf_wmma_f32<<<grid, block, 0, stream>>>(x, y, out);
}